// GNNmodel_1623497638198
// MI455X (gfx1250) — compile-verified
//
#include <hip/hip_runtime.h>
#include <hip/hip_bf16.h>

#define N_NODES 100000
#define N_EDGES 1600000
#define HID     128
#define NH      (N_NODES * HID)       // 12,800,000
#define BN_EPS  1e-5f

typedef float v2f __attribute__((ext_vector_type(2)));
typedef float v8f __attribute__((ext_vector_type(8)));

// ---------------------------------------------------------------- embed ----
__global__ void embed_kernel(const int* __restrict__ x,
                             const float* __restrict__ emb,
                             float* __restrict__ h) {
    int idx = blockIdx.x * blockDim.x + threadIdx.x;
    if (idx >= NH) return;
    int n = idx >> 7, d = idx & 127;
    h[idx] = emb[x[n] * HID + d];
}

// ----------------------------------------------------------------- copy ----
__global__ void copy_kernel(const float* __restrict__ in, float* __restrict__ out) {
    int idx = blockIdx.x * blockDim.x + threadIdx.x;
    if (idx < NH) out[idx] = in[idx];
}

// -------------------------------------------------- scatter (segment_sum) ----
// z[dst[e]] += h[src[e]] ; one thread per (edge, feature)
__global__ void scatter_kernel(const float* __restrict__ h,
                               const long long* __restrict__ src,
                               const long long* __restrict__ dst,
                               float* __restrict__ z) {
    int idx = blockIdx.x * blockDim.x + threadIdx.x;
    if (idx >= N_EDGES * HID) return;       // 204,800,000 < 2^31
    int e = idx >> 7, d = idx & 127;
    int s = (int)src[e], t = (int)dst[e];
    unsafeAtomicAdd(&z[t * HID + d], h[s * HID + d]);
}

// ------------------------------------------------------------ zero stats ----
__global__ void zero_stats_kernel(float* __restrict__ stats) {
    stats[threadIdx.x] = 0.0f;              // 256 threads: sum[128] + sumsq[128]
}

// ------------------------------------------------------------------ GEMM ----
// Y[M x 128] = Z[M x 128] @ W[128 x 128]  via V_WMMA_F32_16X16X4_F32.
// 128 threads = 4 waves; each wave computes a 16-row strip (8 accum tiles).
// W is staged in LDS in *fragment-ready* interleaved layout:
//     Wl[(d>>2)*512 + f*4 + (d&3)] = W[d][f]
// so a lane's B fragment (two consecutive K values of one column) is one
// aligned ds_load_b64 landing directly in an even VGPR pair — no repacking,
// bank-conflict-free (16 B lane stride covers all 64 banks per b64 access).
// Optional: BN column stats (Σv, Σv²) via cross-lane shuffle + f32 atomics;
// optional accumulate (Y += ...), used for the K=512 readout GEMM.
__launch_bounds__(128)
__global__ void gemm_kernel(const float* __restrict__ Z,
                            const float* __restrict__ W,
                            float* __restrict__ Y,
                            float* __restrict__ gstats,
                            int do_stats, int accum, int n_tiles) {
    __shared__ float Wl[HID * HID];         // 64 KB
    {   // cooperative load: coalesced float4 global reads -> interleaved LDS
        const float4* Wv = (const float4*)W;
        for (int i = threadIdx.x; i < HID * HID / 4; i += blockDim.x) {
            float4 w4 = Wv[i];
            int d = i >> 5;                 // row (K index), 0..127
            int f = (i & 31) * 4;           // starting column, 0..124
            int base = (d >> 2) * 512 + (d & 3);
            Wl[base + (f + 0) * 4] = w4.x;
            Wl[base + (f + 1) * 4] = w4.y;
            Wl[base + (f + 2) * 4] = w4.z;
            Wl[base + (f + 3) * 4] = w4.w;
        }
    }
    __syncthreads();

    const int wave = threadIdx.x >> 5;
    const int lane = threadIdx.x & 31;
    const int tile = blockIdx.x * 4 + wave;
    if (tile >= n_tiles) return;

    const int m0     = tile * 16;
    const int laneLo = lane & 15;
    const int kHalf  = (lane >> 4) * 2;     // lanes 0-15: K=0,1 ; 16-31: K=2,3
    const int aRow   = m0 + laneLo;

    v8f acc[8];
    const int rBase = m0 + ((lane >> 4) ? 8 : 0);   // C/D layout: hi lanes -> rows +8
    if (accum) {
        for (int ct = 0; ct < 8; ++ct)
            for (int r = 0; r < 8; ++r)
                acc[ct][r] = Y[(rBase + r) * HID + ct * 16 + laneLo];
    } else {
        for (int ct = 0; ct < 8; ++ct)
            for (int r = 0; r < 8; ++r)
                acc[ct][r] = 0.0f;
    }

    for (int kk = 0; kk < 32; ++kk) {
        // A fragment (16x4 f32): lane holds Z[aRow][4kk+kHalf .. +1] — one b64 load
        v2f a = *(const v2f*)(Z + aRow * HID + kk * 4 + kHalf);
        const float* wbase = Wl + kk * 512 + kHalf;
#pragma unroll
        for (int ct = 0; ct < 8; ++ct) {
            const int col = ct * 16 + laneLo;
            v2f b = *(const v2f*)(wbase + col * 4);   // single ds_load_b64
            acc[ct] = __builtin_amdgcn_wmma_f32_16x16x4_f32(
                false, a, false, b, (short)0, acc[ct], false, false);
        }
    }

#pragma unroll
    for (int ct = 0; ct < 8; ++ct) {
        const int col = ct * 16 + laneLo;
        float s = 0.0f, q = 0.0f;
#pragma unroll
        for (int r = 0; r < 8; ++r) {
            float v = acc[ct][r];
            Y[(rBase + r) * HID + col] = v;
            s += v;
            q += v * v;
        }
        if (do_stats) {
            s += __shfl_xor(s, 16, 32);     // lanes L and L+16 share column `col`
            q += __shfl_xor(q, 16, 32);
            if (lane < 16) {
                unsafeAtomicAdd(&gstats[col], s);
                unsafeAtomicAdd(&gstats[HID + col], q);
            }
        }
    }
}

// -------------------------------------------------- BN finalize (1 block) ----
__global__ void bnfinal_kernel(const float* __restrict__ gstats,
                               const float* __restrict__ gamma,
                               const float* __restrict__ beta,
                               float* __restrict__ scsh) {
    int d = threadIdx.x;                    // 128 threads
    float mean = gstats[d] * (1.0f / N_NODES);
    float var  = gstats[HID + d] * (1.0f / N_NODES) - mean * mean;  // biased
    float inv  = rsqrtf(var + BN_EPS);
    float sc   = gamma[d] * inv;
    scsh[d]       = sc;
    scsh[HID + d] = beta[d] - mean * sc;
}

// -------------------------------------- fused scale/shift + activation ----
// mode 0: relu(v*sc+sh) ; mode 1: elu(v*sc+sh) ; mode 2: relu(v + bias[c])
__global__ void apply_kernel(const float* __restrict__ in,
                             float* __restrict__ out,
                             const float* __restrict__ scsh, int mode) {
    int idx = blockIdx.x * blockDim.x + threadIdx.x;
    if (idx >= NH) return;
    int c = idx & 127;
    float v = in[idx];
    if (mode == 2) {
        v = fmaxf(v + scsh[c], 0.0f);
    } else {
        float t = v * scsh[c] + scsh[HID + c];
        v = (mode == 0) ? fmaxf(t, 0.0f)
                        : (t > 0.0f ? t : (expf(t) - 1.0f));
    }
    out[idx] = v;
}

// ------------------------------------------------------- final GEMV [128->1] ----
__global__ void gemv_kernel(const float* __restrict__ hid,
                            const float* __restrict__ w,
                            const float* __restrict__ b,
                            float* __restrict__ out) {
    int wave = threadIdx.x >> 5, lane = threadIdx.x & 31;
    int n = blockIdx.x * 4 + wave;
    if (n >= N_NODES) return;
    float4 hx = ((const float4*)(hid + n * HID))[lane];
    float4 wx = ((const float4*)w)[lane];
    float s = hx.x * wx.x + hx.y * wx.y + hx.z * wx.z + hx.w * wx.w;
    for (int off = 16; off; off >>= 1) s += __shfl_down(s, off, 32);
    if (lane == 0) out[n] = s + b[0];
}

// ------------------------------------------------------------------ host ----
extern "C" void kernel_launch(void* const* d_in, const int* in_sizes, int n_in,
                              void* d_out, int out_size, void* d_ws, size_t ws_size,
                              hipStream_t stream) {
    const int*       x     = (const int*)d_in[0];
    const long long* ei    = (const long long*)d_in[1];
    const float*     emb   = (const float*)d_in[2];
    const float*     W1    = (const float*)d_in[3];
    const float*     W2    = (const float*)d_in[4];
    const float*     bn1_g = (const float*)d_in[5];
    const float*     bn1_b = (const float*)d_in[6];
    const float*     bn2_g = (const float*)d_in[7];
    const float*     bn2_b = (const float*)d_in[8];
    const float*     rW1   = (const float*)d_in[9];
    const float*     rb1   = (const float*)d_in[10];
    const float*     rW2   = (const float*)d_in[11];
    const float*     rb2   = (const float*)d_in[12];
    float*           out   = (float*)d_out;

    // Workspace: reps R0..R3, tmpA, tmpB (each N*HID f32), stats(256), scsh(256)
    float* ws = (float*)d_ws;
    float* R[4];
    for (int i = 0; i < 4; ++i) R[i] = ws + (size_t)i * NH;
    float* tA    = ws + (size_t)4 * NH;
    float* tB    = ws + (size_t)5 * NH;
    float* stats = ws + (size_t)6 * NH;
    float* scsh  = stats + 256;

    const long long* src = ei;              // edge_index[0]
    const long long* dst = ei + N_EDGES;    // edge_index[1]

    const int TB        = 256;
    const int nBlocksNH = NH / TB;                      // 50000
    const int nBlocksE  = (N_EDGES * HID) / TB;         // 800000
    const int nTiles    = N_NODES / 16;                 // 6250 (exact)
    const int gemmGrid  = (nTiles + 3) / 4;             // 1563

    // h0 = emb[x]
    embed_kernel<<<nBlocksNH, TB, 0, stream>>>(x, emb, R[0]);

    for (int i = 0; i < 3; ++i) {
        // z = h + segment_sum(h[src], dst)
        copy_kernel<<<nBlocksNH, TB, 0, stream>>>(R[i], tA);
        scatter_kernel<<<nBlocksE, TB, 0, stream>>>(R[i], src, dst, tA);
        // z @ W1[i], BN stats
        zero_stats_kernel<<<1, 256, 0, stream>>>(stats);
        gemm_kernel<<<gemmGrid, 128, 0, stream>>>(tA, W1 + i * HID * HID, tB,
                                                  stats, 1, 0, nTiles);
        bnfinal_kernel<<<1, 128, 0, stream>>>(stats, bn1_g + i * HID, bn1_b + i * HID, scsh);
        apply_kernel<<<nBlocksNH, TB, 0, stream>>>(tB, tB, scsh, 0);   // relu(BN)
        // @ W2[i], BN stats
        zero_stats_kernel<<<1, 256, 0, stream>>>(stats);
        gemm_kernel<<<gemmGrid, 128, 0, stream>>>(tB, W2 + i * HID * HID, tA,
                                                  stats, 1, 0, nTiles);
        bnfinal_kernel<<<1, 128, 0, stream>>>(stats, bn2_g + i * HID, bn2_b + i * HID, scsh);
        apply_kernel<<<nBlocksNH, TB, 0, stream>>>(tA, R[i + 1], scsh, 1);  // elu(BN)
    }

    // hid = relu(cat(R0..R3) @ rW1 + rb1)  — K=512 as 4 accumulated K=128 GEMMs
    for (int i = 0; i < 4; ++i)
        gemm_kernel<<<gemmGrid, 128, 0, stream>>>(R[i], rW1 + i * HID * HID, tB,
                                                  nullptr, 0, (i > 0) ? 1 : 0, nTiles);
    apply_kernel<<<nBlocksNH, TB, 0, stream>>>(tB, tB, rb1, 2);

    // out = hid @ rW2 + rb2
    gemv_kernel<<<N_NODES / 4, 128, 0, stream>>>(tB, rW2, rb2, out);
}